// Dynamic_MultiTeacher3_52699248721946
// MI455X (gfx1250) — compile-verified
//
#include <hip/hip_runtime.h>

// Fused multi-teacher KD loss for MI455X (gfx1250, wave32).
// Pass 1: one 256-thread block per row; streams o1,o2,o3,out_s through LDS via
//         CDNA5 async global->LDS loads (double buffered, ASYNCcnt-paced),
//         accumulating sum-exp / weighted-sum / top-2 / target stats per row.
// Pass 2: single 1024-thread block: global max over teachers, per-row loss, mean.

#define LOG2E 1.44269504088896340736f
#define K20   (LOG2E / 20.0f)
#define LN2   0.69314718055994530942f

// ---- fast base-2 transcendentals (v_exp_f32 / v_log_f32) ----
__device__ __forceinline__ float fexp2(float x) {
#if __has_builtin(__builtin_amdgcn_exp2f)
  return __builtin_amdgcn_exp2f(x);
#else
  return exp2f(x);
#endif
}
__device__ __forceinline__ float flog2(float x) {
#if __has_builtin(__builtin_amdgcn_logf)
  return __builtin_amdgcn_logf(x);   // v_log_f32 = log2(x)
#else
  return log2f(x);
#endif
}

// ---- CDNA5 async global->LDS (CDNA5 ISA / CDNA5 ISA op 98) ----
__device__ __forceinline__ void async_ld_b128(unsigned lds_addr, unsigned voff,
                                              const void* sbase) {
#if defined(__gfx1250__)
  // GVS mode: mem = SADDR(64b sgpr pair) + VADDR(32b byte offset) ; VDST = LDS byte addr
  asm volatile("global_load_async_to_lds_b128 %0, %1, %2"
               :: "v"(lds_addr), "v"(voff), "s"(sbase)
               : "memory");
#else
  (void)lds_addr; (void)voff; (void)sbase;
#endif
}
__device__ __forceinline__ void wait_async_le4() {
#if defined(__gfx1250__)
  asm volatile("s_wait_asynccnt 0x4" ::: "memory");
#endif
}
__device__ __forceinline__ void wait_async_le0() {
#if defined(__gfx1250__)
  asm volatile("s_wait_asynccnt 0x0" ::: "memory");
#endif
}

// ---------------------------------------------------------------------------
// Pass 1: per-row statistics.
// stats layout (stride 24 floats per row):
//  [0..3]  Z_t   = sum exp(t/20)              (t = o1,o2,o3,mimic)
//  [4..7]  S_t   = sum exp(t/20) * s
//  [8..11] top1_t
//  [12..15]top2_t
//  [16..19]teacher value at target column
//  [20]    s value at target column
//  [21]    Z1  = sum exp(s)
//  [22]    Z20 = sum exp(s/20)
//  [23]    rowmax over real teachers (for global max_preds)
// ---------------------------------------------------------------------------
__global__ __launch_bounds__(256)
void mt3_pass1(const float* __restrict__ o1, const float* __restrict__ o2,
               const float* __restrict__ o3, const float* __restrict__ os,
               const int* __restrict__ tgt, float* __restrict__ stats,
               int Cn, int nv4, int nch)
{
  const int r   = blockIdx.x;
  const int tid = threadIdx.x;
  const int target = tgt[r];

  const size_t rb = (size_t)r * (size_t)Cn;
  const void* p1 = (const void*)(o1 + rb);
  const void* p2 = (const void*)(o2 + rb);
  const void* p3 = (const void*)(o3 + rb);
  const void* ps = (const void*)(os + rb);

  __shared__ float4 lbuf[2][4][256];   // 32 KB staging (double buffer x 4 arrays)
  __shared__ float  red[18][256];      // 18 KB reduction scratch
  __shared__ float  tg5[5];            // target-column values (one writer)

  float Z[4]  = {0.f, 0.f, 0.f, 0.f};
  float S[4]  = {0.f, 0.f, 0.f, 0.f};
  float T1[4] = {-3.0e38f, -3.0e38f, -3.0e38f, -3.0e38f};
  float T2[4] = {-3.0e38f, -3.0e38f, -3.0e38f, -3.0e38f};
  float Zs1 = 0.f, Zs20 = 0.f;

  auto issue = [&](int c, int buf) {
    int g = c * 256 + tid;
    if (g >= nv4) g = nv4 - 1;                  // clamp tail -> uniform EXEC
    unsigned off = (unsigned)g * 16u;           // byte offset within row
    async_ld_b128((unsigned)(unsigned long long)&lbuf[buf][0][tid], off, p1);
    async_ld_b128((unsigned)(unsigned long long)&lbuf[buf][1][tid], off, p2);
    async_ld_b128((unsigned)(unsigned long long)&lbuf[buf][2][tid], off, p3);
    async_ld_b128((unsigned)(unsigned long long)&lbuf[buf][3][tid], off, ps);
  };

#define TUPD(t_, i_) do { \
    float e_ = fexp2((t_) * K20); \
    Z[i_] += e_; S[i_] = fmaf(e_, sv, S[i_]); \
    if ((t_) > T1[i_]) { T2[i_] = T1[i_]; T1[i_] = (t_); } \
    else               { T2[i_] = fmaxf(T2[i_], (t_)); } \
  } while (0)

#define PELEM(av, bv, cv, sv_, jj) do { \
    float sv = (sv_); \
    float mv = ((av) + (bv) + (cv)) * (1.0f / 3.0f); \
    TUPD((av), 0); TUPD((bv), 1); TUPD((cv), 2); TUPD(mv, 3); \
    Zs1  += fexp2(sv * LOG2E); \
    Zs20 += fexp2(sv * K20); \
    if ((jj) == target) { tg5[0]=(av); tg5[1]=(bv); tg5[2]=(cv); tg5[3]=mv; tg5[4]=sv; } \
  } while (0)

  issue(0, 0);
  for (int c = 0; c < nch; ++c) {
    if (c + 1 < nch) { issue(c + 1, (c + 1) & 1); wait_async_le4(); }
    else             { wait_async_le0(); }

    int g = c * 256 + tid;
    if (g < nv4) {
      int cur = c & 1;
      float4 a  = lbuf[cur][0][tid];
      float4 b  = lbuf[cur][1][tid];
      float4 d  = lbuf[cur][2][tid];
      float4 s4 = lbuf[cur][3][tid];
      int j = g * 4;
      PELEM(a.x, b.x, d.x, s4.x, j + 0);
      PELEM(a.y, b.y, d.y, s4.y, j + 1);
      PELEM(a.z, b.z, d.z, s4.z, j + 2);
      PELEM(a.w, b.w, d.w, s4.w, j + 3);
    }
  }
#undef PELEM
#undef TUPD

  // ---- block reduction (plain sums + top-2 merge) ----
#pragma unroll
  for (int t = 0; t < 4; ++t) {
    red[t][tid] = Z[t]; red[4 + t][tid] = S[t];
    red[8 + t][tid] = T1[t]; red[12 + t][tid] = T2[t];
  }
  red[16][tid] = Zs1; red[17][tid] = Zs20;
  __syncthreads();

  for (int st = 128; st > 0; st >>= 1) {
    if (tid < st) {
#pragma unroll
      for (int t = 0; t < 4; ++t) {
        red[t][tid]     += red[t][tid + st];
        red[4 + t][tid] += red[4 + t][tid + st];
        float a1 = red[8 + t][tid],      a2 = red[12 + t][tid];
        float b1 = red[8 + t][tid + st], b2 = red[12 + t][tid + st];
        if (a1 >= b1) { red[12 + t][tid] = fmaxf(a2, b1); }
        else          { red[8 + t][tid] = b1; red[12 + t][tid] = fmaxf(b2, a1); }
      }
      red[16][tid] += red[16][tid + st];
      red[17][tid] += red[17][tid + st];
    }
    __syncthreads();
  }

  if (tid == 0) {
    float* w = stats + (size_t)r * 24;
#pragma unroll
    for (int t = 0; t < 4; ++t) {
      w[t] = red[t][0]; w[4 + t] = red[4 + t][0];
      w[8 + t] = red[8 + t][0]; w[12 + t] = red[12 + t][0];
      w[16 + t] = tg5[t];
    }
    w[20] = tg5[4];
    w[21] = red[16][0];
    w[22] = red[17][0];
    w[23] = fmaxf(fmaxf(red[8][0], red[9][0]), red[10][0]);
  }
}

// ---------------------------------------------------------------------------
// Pass 2: global max_preds, per-row loss, mean.  One block, 1024 threads.
// ---------------------------------------------------------------------------
__global__ __launch_bounds__(1024)
void mt3_pass2(const float* __restrict__ stats, float* __restrict__ out, int Bn)
{
  __shared__ float sh[1024];
  const int tid = threadIdx.x;

  float m = -3.0e38f;
  for (int r = tid; r < Bn; r += 1024) m = fmaxf(m, stats[(size_t)r * 24 + 23]);
  sh[tid] = m; __syncthreads();
  for (int s = 512; s > 0; s >>= 1) {
    if (tid < s) sh[tid] = fmaxf(sh[tid], sh[tid + s]);
    __syncthreads();
  }
  const float maxpreds = sh[0];
  __syncthreads();

  float acc = 0.0f;
  for (int r = tid; r < Bn; r += 1024) {
    const float* w = stats + (size_t)r * 24;
    float lse1  = LN2 * flog2(w[21]);   // logsumexp(s)
    float lse20 = LN2 * flog2(w[22]);   // logsumexp(s/20)
    float ce = lse1 - w[20];

    float dv[4], ls[4];
#pragma unroll
    for (int t = 0; t < 4; ++t) {
      float Eqs = w[4 + t] / w[t];                      // E_q[s]
      float kd  = 400.0f * lse20 - 20.0f * Eqs;         // T^2 * KL cross term
      float tv  = w[16 + t];
      dv[t] = (tv == w[8 + t]) ? (w[8 + t] - w[12 + t]) : 0.0f;
      float wt = 0.8f * tv / maxpreds;
      ls[t] = (1.0f - wt) * ce + wt * kd;
    }
    float dm = fmaxf(fmaxf(dv[0], dv[1]), fmaxf(dv[2], dv[3]));
    float es = 0.0f, rl = 0.0f;
#pragma unroll
    for (int t = 0; t < 4; ++t) {
      float e = fexp2((dv[t] - dm) * (0.5f * LOG2E)); // softmax(d / 2)
      es += e; rl = fmaf(e, ls[t], rl);
    }
    acc += rl / es;
  }

  sh[tid] = acc; __syncthreads();
  for (int s = 512; s > 0; s >>= 1) {
    if (tid < s) sh[tid] += sh[tid + s];
    __syncthreads();
  }
  if (tid == 0) out[0] = sh[0] / (float)Bn;
}

// ---------------------------------------------------------------------------
extern "C" void kernel_launch(void* const* d_in, const int* in_sizes, int n_in,
                              void* d_out, int out_size, void* d_ws, size_t ws_size,
                              hipStream_t stream) {
  const float* o1 = (const float*)d_in[0];
  const float* o2 = (const float*)d_in[1];
  const float* o3 = (const float*)d_in[2];
  const float* os = (const float*)d_in[3];
  const int*   tg = (const int*)d_in[4];

  const int Bn  = in_sizes[4];              // 4096
  const int Cn  = in_sizes[0] / Bn;         // 10000 (divisible by 4)
  const int nv4 = Cn / 4;                   // float4 slots per row
  const int nch = (nv4 + 255) / 256;        // chunks of 1024 elements

  float* stats = (float*)d_ws;              // needs Bn * 24 * 4 bytes (~384 KB)

  mt3_pass1<<<Bn, 256, 0, stream>>>(o1, o2, o3, os, tg, stats, Cn, nv4, nch);
  mt3_pass2<<<1, 1024, 0, stream>>>(stats, (float*)d_out, Bn);
}